// PinballLoss_43121471651913
// MI455X (gfx1250) — compile-verified
//
#include <hip/hip_runtime.h>
#include <hip/hip_bf16.h>

// Pinball (quantile) loss -> single scalar mean.
// Bandwidth-bound streaming reduction: ~151 MB read once @ 23.3 TB/s floor.
// Pass 1: grid-stride NT b128 loads, per-lane fp32 accumulate, wave reduce via
//         V_WMMA_F32_16X16X4_F32 (B = ones), block partials -> d_ws.
// Pass 2: single block reduces the 1024 partials and writes mean to d_out.

typedef __attribute__((ext_vector_type(4))) float v4f;
typedef __attribute__((ext_vector_type(2))) float v2f;
typedef __attribute__((ext_vector_type(8))) float v8f;

#define NBLOCKS  1024
#define NTHREADS 256

__global__ __launch_bounds__(NTHREADS) void pinball_partial_kernel(
    const float* __restrict__ preds,
    const float* __restrict__ target,
    const float* __restrict__ quantiles,
    float* __restrict__ partial,
    int n)
{
    const float q   = quantiles[0];   // scalar load (uniform)
    const float qm1 = q - 1.0f;

    const int n4     = n >> 2;
    const int gid    = blockIdx.x * NTHREADS + (int)threadIdx.x;
    const int stride = NBLOCKS * NTHREADS;

    const v4f* __restrict__ p4 = (const v4f*)preds;
    const v4f* __restrict__ t4 = (const v4f*)target;

    float acc = 0.0f;
    for (int i = gid; i < n4; i += stride) {
        v4f p = __builtin_nontemporal_load(p4 + i);   // global_load_b128, TH=NT
        v4f t = __builtin_nontemporal_load(t4 + i);
        v4f e = t - p;
        acc += fmaxf(qm1 * e.x, q * e.x);
        acc += fmaxf(qm1 * e.y, q * e.y);
        acc += fmaxf(qm1 * e.z, q * e.z);
        acc += fmaxf(qm1 * e.w, q * e.w);
    }
    // Scalar tail (n % 4 != 0 safety); reconverges before the WMMA below.
    if (gid == 0) {
        for (int i = (n4 << 2); i < n; ++i) {
            float e = target[i] - preds[i];
            acc += fmaxf(qm1 * e, q * e);
        }
    }

    // ---- wave32 reduction via V_WMMA_F32_16X16X4_F32 ----
    // A (16x4 f32): VGPR0 = per-lane acc (lanes 0..15 -> K=0, lanes 16..31 -> K=2),
    //               VGPR1 = 0 (K=1,3). B = all ones -> D[m][n] = acc[m] + acc[m+16].
    v2f a; a.x = acc;  a.y = 0.0f;
    v2f b; b.x = 1.0f; b.y = 1.0f;
    v8f c = {};
    c = __builtin_amdgcn_wmma_f32_16x16x4_f32(
        /*neg_a=*/false, a, /*neg_b=*/false, b,
        /*c_mod=*/(short)0, c, /*reuse_a=*/false, /*reuse_b=*/false);

    // Per the 16x16 f32 C/D layout: lane holds rows {v} (lanes<16) / {v+8} (lanes>=16)
    // of one column across VGPRs 0..7. Sum them, then fold the two halves.
    float s = c[0] + c[1] + c[2] + c[3] + c[4] + c[5] + c[6] + c[7];
    s += __shfl_xor(s, 16, 32);   // rows 0..7 + rows 8..15 -> wave total (all lanes)

    __shared__ float lsum[NTHREADS / 32];
    const int lane = threadIdx.x & 31;
    const int wave = threadIdx.x >> 5;
    if (lane == 0) lsum[wave] = s;
    __syncthreads();
    if (threadIdx.x == 0) {
        float bs = 0.0f;
        #pragma unroll
        for (int w = 0; w < NTHREADS / 32; ++w) bs += lsum[w];
        partial[blockIdx.x] = bs;   // one slot per block: deterministic, no atomics
    }
}

__global__ __launch_bounds__(NTHREADS) void pinball_final_kernel(
    const float* __restrict__ partial,
    float* __restrict__ out,
    int n)
{
    __shared__ float lsum[NTHREADS];
    float s = 0.0f;
    for (int i = threadIdx.x; i < NBLOCKS; i += NTHREADS) s += partial[i];
    lsum[threadIdx.x] = s;
    __syncthreads();
    for (int off = NTHREADS / 2; off > 0; off >>= 1) {
        if ((int)threadIdx.x < off) lsum[threadIdx.x] += lsum[threadIdx.x + off];
        __syncthreads();
    }
    if (threadIdx.x == 0) out[0] = lsum[0] / (float)n;
}

extern "C" void kernel_launch(void* const* d_in, const int* in_sizes, int n_in,
                              void* d_out, int out_size, void* d_ws, size_t ws_size,
                              hipStream_t stream) {
    const float* preds     = (const float*)d_in[0];
    const float* target    = (const float*)d_in[1];
    const float* quantiles = (const float*)d_in[2];
    float* out     = (float*)d_out;
    float* partial = (float*)d_ws;          // NBLOCKS floats = 4 KB scratch
    const int n = in_sizes[0];              // B*N*T elements

    pinball_partial_kernel<<<NBLOCKS, NTHREADS, 0, stream>>>(preds, target, quantiles, partial, n);
    pinball_final_kernel<<<1, NTHREADS, 0, stream>>>(partial, out, n);
}